// LPKTPlus_37546604102383
// MI455X (gfx1250) — compile-verified
//
#include <hip/hip_runtime.h>
#include <hip/hip_bf16.h>
#include <math.h>

// ---------------------------------------------------------------------------
// LPKT recurrence for MI455X (gfx1250, wave32).
// One workgroup per batch element. State h[128][64] lives in LDS (stride 66
// to avoid bank conflicts). Per-step GEMM h_pre @ W4a^T done with
// V_WMMA_F32_16X16X4_F32; W4a^T B-fragments are preloaded in VGPRs.
// ---------------------------------------------------------------------------

typedef __attribute__((ext_vector_type(2))) float v2f;
typedef __attribute__((ext_vector_type(8))) float v8f;

#define BQ   32      // batch
#define LQ   200     // seq len
#define NCQ  128     // concepts
#define DKQ  64      // hidden
#define HS   66      // padded LDS row stride for h (floats)

__device__ __forceinline__ float sigmoidf_(float x) {
    return 1.0f / (1.0f + __expf(-x));
}

__global__ __launch_bounds__(256)
void lpkt_scan_kernel(const int*   __restrict__ question_seq,   // [B,L]
                      const int*   __restrict__ interval_seq,   // [B,L]
                      const int*   __restrict__ usetime_seq,    // [B,L]
                      const float* __restrict__ correct_seq,    // [B,L]
                      const float* __restrict__ q_matrix,       // [NQ+1,128]
                      const float* __restrict__ E_q,            // [NQ+1,64]
                      const float* __restrict__ E_at,           // [3601,64]
                      const float* __restrict__ E_it,           // [43201,64]
                      const float* __restrict__ W1,             // [64,178]
                      const float* __restrict__ b1,             // [64]
                      const float* __restrict__ W2,             // [64,256]
                      const float* __restrict__ b2,
                      const float* __restrict__ W3,             // [64,256]
                      const float* __restrict__ b3,
                      const float* __restrict__ W4,             // [64,192]
                      const float* __restrict__ b4,
                      const float* __restrict__ W5,             // [64,128]
                      const float* __restrict__ b5,
                      const float* __restrict__ h0,             // [128,64]
                      float*       __restrict__ pred)           // [B,L]
{
    __shared__ float sh_h[NCQ * HS];        // state h (padded rows)
    __shared__ float sh_qet[NCQ];           // q_matrix row at t
    __shared__ float sh_qen[NCQ];           // q_matrix row at t+1
    __shared__ float sh_it[DKQ];            // interval-time embedding at t
    __shared__ float sh_qembt[DKQ];         // E_q row at t
    __shared__ float sh_atemb[DKQ];         // E_at row at t
    __shared__ float sh_qembn[DKQ];         // E_q row at t+1
    __shared__ float sh_lrn[DKQ];           // all_learning at t
    __shared__ float sh_lrnpre[DKQ];        // all_learning at t-1
    __shared__ float sh_htld[DKQ];          // h_tilde carry
    __shared__ float sh_LG[DKQ];
    __shared__ float sh_rvec[DKQ];          // broadcast part of gamma_f pre-act
    __shared__ float sh_w1csum[DKQ];        // sum over correctness cols of W1
    __shared__ float sh_g[4 * DKQ];         // g_in (256)
    __shared__ float sh_p2[DKQ * 4];        // matvec partials for W2
    __shared__ float sh_p3[DKQ * 4];        // matvec partials for W3
    __shared__ float sh_y[DKQ];

    const int b    = blockIdx.x;
    const int tid  = threadIdx.x;
    const int wave = tid >> 5;
    const int lane = tid & 31;
    const int lhi  = lane >> 4;      // 0 for lanes 0-15, 1 for 16-31
    const int llo  = lane & 15;

    // ---- Tile assignment: wave -> nt = wave&3, mt = (wave>>2) + 2*i ----
    const int nt  = wave & 3;
    const int mt0 = wave >> 2;

    // ---- Preload loop-invariant B fragments of W4a^T (B[k][n] = W4[n][k]) ----
    v2f bfrag[16];
#pragma unroll
    for (int kb = 0; kb < 16; ++kb) {
        const int n  = nt * 16 + llo;
        const int k0 = kb * 4 + 2 * lhi;
        bfrag[kb].x = W4[n * 192 + k0];
        bfrag[kb].y = W4[n * 192 + k0 + 1];
    }

    // ---- W1 correctness-column sums ----
    if (tid < DKQ) {
        float s = 0.f;
        for (int k = 128; k < 178; ++k) s += W1[tid * 178 + k];
        sh_w1csum[tid] = s;
    }

    // ---- h <- h0 (broadcast over batch) ----
    for (int i = tid; i < NCQ * DKQ; i += 256)
        sh_h[(i >> 6) * HS + (i & 63)] = h0[i];

    // ---- qe row for t=0 and initial carries ----
    {
        const int q0 = question_seq[b * LQ + 0];
        if (tid < NCQ) sh_qet[tid] = q_matrix[q0 * NCQ + tid];
        if (tid < DKQ) sh_lrnpre[tid] = 0.f;
        if (tid == 0)  pred[b * LQ + 0] = 0.f;
    }
    __syncthreads();
    if (tid < DKQ) {                       // h_tilde0 = qe0 . h0
        float s = 0.f;
        for (int c = 0; c < NCQ; ++c) s += sh_qet[c] * sh_h[c * HS + tid];
        sh_htld[tid] = s;
    }
    __syncthreads();

    // =======================  recurrence over t  =========================
    for (int t = 0; t < LQ - 1; ++t) {
        // -- load step inputs --
        const int qid_t = question_seq[b * LQ + t];
        const int qid_n = question_seq[b * LQ + t + 1];
        const int itid  = interval_seq[b * LQ + t];
        const int atid  = usetime_seq[b * LQ + t];
        const float corr = correct_seq[b * LQ + t];

        if (tid < NCQ)                 sh_qet[tid]        = q_matrix[qid_t * NCQ + tid];
        else                           sh_qen[tid - NCQ]  = q_matrix[qid_n * NCQ + (tid - NCQ)];
        if (tid < DKQ)                 sh_it[tid]         = E_it[itid * DKQ + tid];
        else if (tid < 2 * DKQ)        sh_qembt[tid - 64] = E_q[qid_t * DKQ + (tid - 64)];
        else if (tid < 3 * DKQ)        sh_atemb[tid - 128]= E_at[atid * DKQ + (tid - 128)];
        else                           sh_qembn[tid - 192]= E_q[qid_n * DKQ + (tid - 192)];
        __syncthreads();

        // -- all_learning[t] --
        if (tid < DKQ) {
            float s = b1[tid] + corr * sh_w1csum[tid];
            const float* w = &W1[tid * 178];
            for (int k = 0; k < DKQ; ++k) s += sh_qembt[k] * w[k];
            for (int k = 0; k < DKQ; ++k) s += sh_atemb[k] * w[64 + k];
            sh_lrn[tid] = s;
        }
        __syncthreads();

        // -- assemble g_in = [lrn_pre | it | lrn | h_tilde_pre] --
        {
            const int blk = tid >> 6, off = tid & 63;
            float v = (blk == 0) ? sh_lrnpre[off]
                    : (blk == 1) ? sh_it[off]
                    : (blk == 2) ? sh_lrn[off]
                                 : sh_htld[off];
            sh_g[tid] = v;
        }
        __syncthreads();

        // -- W2/W3 matvec partials (4 threads per output) --
        {
            const int j = tid >> 2, p = tid & 3;
            const float* w2 = &W2[j * 256 + p * 64];
            const float* w3 = &W3[j * 256 + p * 64];
            const float* g  = &sh_g[p * 64];
            float s2 = 0.f, s3 = 0.f;
            for (int k = 0; k < 64; ++k) { s2 += g[k] * w2[k]; s3 += g[k] * w3[k]; }
            sh_p2[tid] = s2; sh_p3[tid] = s3;
        }
        __syncthreads();

        // -- LG, and carry lrn_pre <- lrn in parallel --
        if (tid < DKQ) {
            float s2 = sh_p2[tid*4] + sh_p2[tid*4+1] + sh_p2[tid*4+2] + sh_p2[tid*4+3] + b2[tid];
            float s3 = sh_p3[tid*4] + sh_p3[tid*4+1] + sh_p3[tid*4+2] + sh_p3[tid*4+3] + b3[tid];
            float lg = tanhf(s2);
            float gl = sigmoidf_(s3);
            sh_LG[tid] = gl * (lg + 1.0f) * 0.5f;
        } else if (tid < 2 * DKQ) {
            sh_lrnpre[tid - 64] = sh_lrn[tid - 64];
        }
        __syncthreads();

        // -- rvec[j] = b4[j] + LG.W4[:,64:128] + it.W4[:,128:192] --
        if (tid < DKQ) {
            float s = b4[tid];
            const float* w = &W4[tid * 192];
            for (int k = 0; k < DKQ; ++k) s += sh_LG[k] * w[64 + k];
            for (int k = 0; k < DKQ; ++k) s += sh_it[k] * w[128 + k];
            sh_rvec[tid] = s;
        }
        __syncthreads();

        // -- WMMA: acc = h_pre[128,64] @ W4a^T[64,64], tile (mt,nt) per wave --
        v8f acc[4];
#pragma unroll
        for (int i = 0; i < 4; ++i) acc[i] = (v8f){0.f,0.f,0.f,0.f,0.f,0.f,0.f,0.f};

#pragma unroll
        for (int kb = 0; kb < 16; ++kb) {
#pragma unroll
            for (int i = 0; i < 4; ++i) {
                const int m = (mt0 + 2 * i) * 16 + llo;
                const v2f a = *(const v2f*)&sh_h[m * HS + kb * 4 + 2 * lhi];
                acc[i] = __builtin_amdgcn_wmma_f32_16x16x4_f32(
                            false, a, false, bfrag[kb], (short)0, acc[i],
                            false, false);
            }
        }
        __syncthreads();   // all WMMA reads of h done before anyone writes h

        // -- state update: h = qe_t (x) LG + sigmoid(acc + rvec) * h_pre --
#pragma unroll
        for (int i = 0; i < 4; ++i) {
            const int mt  = mt0 + 2 * i;
            const int col = nt * 16 + llo;
            const float lgc = sh_LG[col];
            const float rvc = sh_rvec[col];
#pragma unroll
            for (int v = 0; v < 8; ++v) {
                const int row = mt * 16 + v + 8 * lhi;
                const float gf = sigmoidf_(acc[i][v] + rvc);
                const float hp = sh_h[row * HS + col];
                sh_h[row * HS + col] = sh_qet[row] * lgc + gf * hp;
            }
        }
        __syncthreads();

        // -- h_tilde = qe_next . h --
        if (tid < DKQ) {
            float s = 0.f;
            for (int c = 0; c < NCQ; ++c) s += sh_qen[c] * sh_h[c * HS + tid];
            sh_htld[tid] = s;
        }
        __syncthreads();

        // -- y = mean_j sigmoid([q_emb_next | h_tilde] . W5[j] + b5[j]) --
        if (tid < DKQ) {
            float s = b5[tid];
            const float* w = &W5[tid * 128];
            for (int k = 0; k < DKQ; ++k) s += sh_qembn[k] * w[k];
            for (int k = 0; k < DKQ; ++k) s += sh_htld[k]  * w[64 + k];
            sh_y[tid] = sigmoidf_(s);
        }
        __syncthreads();
        if (tid == 0) {
            float s = 0.f;
            for (int j = 0; j < DKQ; ++j) s += sh_y[j];
            pred[b * LQ + t + 1] = s * (1.0f / 64.0f);
        }
        __syncthreads();   // protect buffers before next step's loads
    }
}

extern "C" void kernel_launch(void* const* d_in, const int* in_sizes, int n_in,
                              void* d_out, int out_size, void* d_ws, size_t ws_size,
                              hipStream_t stream) {
    (void)in_sizes; (void)n_in; (void)d_ws; (void)ws_size; (void)out_size;
    const int*   question_seq = (const int*)  d_in[0];
    const int*   usetime_seq  = (const int*)  d_in[1];
    const int*   interval_seq = (const int*)  d_in[2];
    const float* correct_seq  = (const float*)d_in[3];
    const float* q_matrix     = (const float*)d_in[4];
    const float* E_q          = (const float*)d_in[5];
    const float* E_at         = (const float*)d_in[6];
    const float* E_it         = (const float*)d_in[7];
    const float* W1 = (const float*)d_in[8];
    const float* b1 = (const float*)d_in[9];
    const float* W2 = (const float*)d_in[10];
    const float* b2 = (const float*)d_in[11];
    const float* W3 = (const float*)d_in[12];
    const float* b3 = (const float*)d_in[13];
    const float* W4 = (const float*)d_in[14];
    const float* b4 = (const float*)d_in[15];
    const float* W5 = (const float*)d_in[16];
    const float* b5 = (const float*)d_in[17];
    const float* h0 = (const float*)d_in[18];
    float* pred = (float*)d_out;

    lpkt_scan_kernel<<<BQ, 256, 0, stream>>>(
        question_seq, interval_seq, usetime_seq, correct_seq,
        q_matrix, E_q, E_at, E_it,
        W1, b1, W2, b2, W3, b3, W4, b4, W5, b5, h0, pred);
}